// WavePDE_45389214384298
// MI455X (gfx1250) — compile-verified
//
#include <hip/hip_runtime.h>
#include <cstdint>

// ---------------------------------------------------------------------------
// WavePDE fused pipeline for MI455X (gfx1250, wave32, WMMA).
//   - all big GEMMs (2048x512x512 and the 2048x2048x512 JVP) via
//     V_WMMA_F32_16X16X4_F32 (f32 keeps double-backprop precision; workload
//     is L2-resident and latency-bound, so f32 WMMA costs nothing at roofline)
//   - weight tiles staged with GLOBAL_LOAD_ASYNC_TO_LDS (ASYNCcnt DMA path)
//     when the builtins are available; synchronous fallback otherwise
//   - u_zz via hand-derived double backprop (3 extra GEMMs/step)
//   - u_tt via scalar forward-mode through the time MLP (one small kernel)
//   - scalars (mse_ic / mse_pde / mse_jvp / u_tt) accumulated with f32 atomics
// Fragment layouts per CDNA5 ISA 7.12.2 (C/D: lane%16=N, vgpr+8*(lane/16)=M;
// A 16x4: lane%16=M, k=2*(lane/16)+vgpr; B assumed mirror of A with N for M).
// ---------------------------------------------------------------------------

typedef float v8f __attribute__((ext_vector_type(8)));
typedef float v2f __attribute__((ext_vector_type(2)));
typedef int   v4i_g __attribute__((vector_size(16)));   // matches builtin proto

#define B_   2048
#define D_   512
#define DG_  2048
#define BM   64
#define BN   64
#define BK   32
#define LDT  (BK + 4)   // 36 floats: row base 144B -> 16B aligned (b128 LDS), even k -> 8B

#define AS1 __attribute__((address_space(1)))
#define AS3 __attribute__((address_space(3)))

#if defined(__has_builtin)
#if __has_builtin(__builtin_amdgcn_global_load_async_to_lds_b128) && \
    __has_builtin(__builtin_amdgcn_global_load_async_to_lds_b32)  && \
    __has_builtin(__builtin_amdgcn_s_wait_asynccnt)
#define USE_ASYNC_LDS 1
#endif
#endif
#ifndef USE_ASYNC_LDS
#define USE_ASYNC_LDS 0
#endif

struct GP {
  const float* Ain;   // A matrix source (row-major, ld = D_)
  const float* W;     // weight matrix (row-major, ld = D_)
  const float* bias;  // bx / bf
  const float* te;    // time embedding for this step (D_)
  const float* x;     // activations
  const float* h;
  const float* dx;
  const float* q;
  const float* r3;    // per-row 1-u^2
  const float* u;     // per-row u
  const float* psi;   // per-row psi
  const float* Wo;
  const float* wxs;   // row-sums of Wx
  const float* zcur;  // current latent
  const float* uz;    // u_z buffer
  const float* cptr;  // c parameter
  const int*   idxp;  // index scalar
  const int*   tselp; // t_sel scalar
  const float* scal;  // scalar block (utt[4], ic, pde, jvp)
  float* Out;         // main matrix output
  float* znext;       // next latent
  float* lat1;        // d_out latent1
  float* lat2;        // d_out latent2
  float* accum;       // scalar accumulation target
  int step;
  int N;              // output columns (D_ or DG_)
};

// A-operand fusion per GEMM mode.
// 0: s1 = z            1: s2 A = x + te         2: d_s2 = r3*Wo*(1-h^2)
// 3: d_s1 = dx*(1-x^2) 4: v = (1-x^2)*wxs      5: alpha = -2h*Wo*q*(1-h^2)
// 6: gamma (plain)     7: u_z (plain, JVP)
template <int MODE>
__device__ inline float aprep(const GP& p, int r, int c, float g) {
  if constexpr (MODE == 0 || MODE == 6 || MODE == 7) {
    return g;
  } else if constexpr (MODE == 1) {
    return g + p.te[c];
  } else if constexpr (MODE == 2) {
    return p.r3[r] * p.Wo[c] * (1.f - g * g);
  } else if constexpr (MODE == 3) {
    float xx = p.x[(size_t)r * D_ + c];
    return g * (1.f - xx * xx);
  } else if constexpr (MODE == 4) {
    return (1.f - g * g) * p.wxs[c];
  } else {  // MODE == 5
    float qv = p.q[(size_t)r * D_ + c];
    return -2.f * g * p.Wo[c] * qv * (1.f - g * g);
  }
}

template <int MODE>
__global__ __launch_bounds__(256) void gemm_k(GP p) {
  // JVP only runs at step == t_sel+1 (t_sel is a device scalar).
  if constexpr (MODE == 7) {
    if (p.tselp[0] + 1 != p.step) return;
  }
  constexpr bool FWD = (MODE == 0 || MODE == 1 || MODE == 4 || MODE == 7);  // C = A @ W^T
  constexpr bool RED = (MODE == 3 || MODE == 6 || MODE == 7);

  __shared__ float Al[BM * LDT];
  __shared__ float Bl[BN * LDT];
  __shared__ float red[256];

  const int tid  = threadIdx.x;
  const int lane = tid & 31;
  const int wv   = tid >> 5;
  const int rb   = blockIdx.y * BM;
  const int cb   = blockIdx.x * BN;
  const int m0   = (wv & 3) * 16;   // wave row offset in macro tile
  const int n0   = (wv >> 2) * 32;  // wave col offset (two 16-wide subtiles)

  v8f accs[2];
  const v8f z8 = {0.f, 0.f, 0.f, 0.f, 0.f, 0.f, 0.f, 0.f};
  accs[0] = z8;
  accs[1] = z8;

  const int sr  = tid >> 2;         // 0..63 staging row
  const int sc4 = (tid & 3) * 4;    // 0,4,8,12 staging col group base

  for (int kb = 0; kb < D_; kb += BK) {
    // ---- stage B as Bt[n][k] (pure copy -> async DMA to LDS) ----
    if constexpr (FWD) {
#pragma unroll
      for (int half = 0; half < 2; ++half) {
        const int col = sc4 + half * 16;
        float* src = const_cast<float*>(p.W + (size_t)(cb + sr) * D_ + kb + col);
#if USE_ASYNC_LDS
        __builtin_amdgcn_global_load_async_to_lds_b128(
            (AS1 v4i_g*)src, (AS3 v4i_g*)&Bl[sr * LDT + col], 0, 0);
#else
        float4 g = *(const float4*)src;
        Bl[sr * LDT + col + 0] = g.x;
        Bl[sr * LDT + col + 1] = g.y;
        Bl[sr * LDT + col + 2] = g.z;
        Bl[sr * LDT + col + 3] = g.w;
#endif
      }
    } else {
      const int n  = tid & 63;
      const int k0 = tid >> 6;  // 0..3
#pragma unroll
      for (int pp = 0; pp < 8; ++pp) {
        const int k = k0 + 4 * pp;  // covers 0..31
        float* src = const_cast<float*>(p.W + (size_t)(kb + k) * D_ + cb + n);
#if USE_ASYNC_LDS
        __builtin_amdgcn_global_load_async_to_lds_b32(
            (AS1 int*)src, (AS3 int*)&Bl[n * LDT + k], 0, 0);
#else
        Bl[n * LDT + k] = *src;
#endif
      }
    }
    // ---- stage A (with fused elementwise transform; needs VGPR round-trip) ----
#pragma unroll
    for (int half = 0; half < 2; ++half) {
      const int col = sc4 + half * 16;
      const float* src = p.Ain + (size_t)(rb + sr) * D_ + kb + col;
      float4 g = *(const float4*)src;
      Al[sr * LDT + col + 0] = aprep<MODE>(p, rb + sr, kb + col + 0, g.x);
      Al[sr * LDT + col + 1] = aprep<MODE>(p, rb + sr, kb + col + 1, g.y);
      Al[sr * LDT + col + 2] = aprep<MODE>(p, rb + sr, kb + col + 2, g.z);
      Al[sr * LDT + col + 3] = aprep<MODE>(p, rb + sr, kb + col + 3, g.w);
    }
    // prefetch next A tile (global_prefetch_b8)
    if (kb + BK < D_) {
      __builtin_prefetch(p.Ain + (size_t)(rb + sr) * D_ + kb + BK + sc4, 0, 1);
    }
#if USE_ASYNC_LDS
    __builtin_amdgcn_s_wait_asynccnt(0);
#endif
    __syncthreads();

    // ---- 8 k-steps x 2 subtiles of WMMA_F32_16X16X4_F32 ----
#pragma unroll
    for (int ks = 0; ks < BK; ks += 4) {
      const int kk = ks + 2 * (lane >> 4);
      const int am = m0 + (lane & 15);
      v2f a;
      a.x = Al[am * LDT + kk];
      a.y = Al[am * LDT + kk + 1];
      {
        const int bn = n0 + (lane & 15);
        v2f b;
        b.x = Bl[bn * LDT + kk];
        b.y = Bl[bn * LDT + kk + 1];
        accs[0] = __builtin_amdgcn_wmma_f32_16x16x4_f32(false, a, false, b,
                                                        (short)0, accs[0], false, false);
      }
      {
        const int bn = n0 + 16 + (lane & 15);
        v2f b;
        b.x = Bl[bn * LDT + kk];
        b.y = Bl[bn * LDT + kk + 1];
        accs[1] = __builtin_amdgcn_wmma_f32_16x16x4_f32(false, a, false, b,
                                                        (short)0, accs[1], false, false);
      }
    }
    __syncthreads();
  }

  // ---- epilogue ----
  float part = 0.f;
  const int rloc = m0 + 8 * (lane >> 4);
  const int cl0  = lane & 15;
#pragma unroll
  for (int s = 0; s < 2; ++s) {
#pragma unroll
    for (int v = 0; v < 8; ++v) {
      const int r = rb + rloc + v;
      const int c = cb + n0 + s * 16 + cl0;
      const size_t idx = (size_t)r * D_ + c;
      float val = accs[s][v];
      if constexpr (MODE == 0 || MODE == 1) {
        p.Out[idx] = tanhf(val + p.bias[c]);           // x or h
      } else if constexpr (MODE == 2 || MODE == 4) {
        p.Out[idx] = val;                              // d_x or q
      } else if constexpr (MODE == 3) {
        float uzv = val;
        p.Out[idx] = uzv;                              // u_z
        float zn = p.zcur[idx] + uzv;
        p.znext[idx] = zn;
        int ts = p.tselp[0];
        if (p.step == ts)     p.lat1[idx] = zn;        // latent1
        if (p.step == ts + 1) p.lat2[idx] = zn;        // latent2
        if (p.step == 0) part += uzv * uzv;            // mse_ic (t=0 reuse)
      } else if constexpr (MODE == 5) {
        float xx = p.x[idx];
        float dd = p.dx[idx];
        float g = (val - 2.f * xx * p.wxs[c] * dd / p.r3[r]) * (1.f - xx * xx);
        p.Out[idx] = g;                                // gamma
      } else if constexpr (MODE == 6) {
        float uzz = p.r3[r] * val - 2.f * p.u[r] * p.psi[r] * p.uz[idx];
        float c0 = p.cptr[p.idxp[0]];
        float pd = p.scal[p.step] - c0 * c0 * uzz;     // u_tt - c0^2 u_zz
        part += pd * pd;
      } else {  // MODE == 7 (JVP)
        part += val * val;
      }
    }
  }

  if constexpr (RED) {
    bool doit = (MODE != 3) || (p.step == 0);
    if (doit) {
      red[tid] = part;
      __syncthreads();
      for (int o = 128; o > 0; o >>= 1) {
        if (tid < o) red[tid] += red[tid + o];
        __syncthreads();
      }
      if (tid == 0) atomicAdd(p.accum, red[0]);
    }
  }
}

// Per-row reductions: u = tanh(Wo.h + bo), r3, and forward-mode u_tt terms.
__global__ __launch_bounds__(256) void r1_k(const float* h, const float* Wo,
                                            const float* bo, const float* S2d,
                                            const float* S2dd, float* u, float* r3,
                                            float* uttslot, float* energy,
                                            const int* tselp, int step) {
  const int lane = threadIdx.x & 31;
  const int wv = threadIdx.x >> 5;
  const int row = blockIdx.x * 8 + wv;
  float s3 = 0.f, sd = 0.f, sdd = 0.f;
  for (int j = lane; j < D_; j += 32) {
    float hv = h[(size_t)row * D_ + j];
    float wo = Wo[j];
    float oh = 1.f - hv * hv;
    float a = S2d[j], b = S2dd[j];
    s3 += wo * hv;
    sd += wo * oh * a;                            // Wo . h_dot
    sdd += wo * (oh * b - 2.f * hv * oh * a * a); // Wo . h_ddot
  }
  for (int o = 16; o > 0; o >>= 1) {
    s3 += __shfl_xor(s3, o, 32);
    sd += __shfl_xor(sd, o, 32);
    sdd += __shfl_xor(sdd, o, 32);
  }
  if (lane == 0) {
    float uu = tanhf(s3 + bo[0]);
    float rr = 1.f - uu * uu;
    float udd = rr * sdd - 2.f * uu * rr * sd * sd;  // u_ddot per sample
    u[row] = uu;
    r3[row] = rr;
    atomicAdd(uttslot, udd);                          // u_tt = sum_b u_ddot
    if (step == tselp[0]) energy[row] = uu;           // energy output
  }
}

// psi_b = sum_j Wo[j]*(1-h^2)*q[b,j]
__global__ __launch_bounds__(256) void r2_k(const float* h, const float* Wo,
                                            const float* q, float* psi) {
  const int lane = threadIdx.x & 31;
  const int wv = threadIdx.x >> 5;
  const int row = blockIdx.x * 8 + wv;
  float s = 0.f;
  for (int j = lane; j < D_; j += 32) {
    float hv = h[(size_t)row * D_ + j];
    s += Wo[j] * (1.f - hv * hv) * q[(size_t)row * D_ + j];
  }
  for (int o = 16; o > 0; o >>= 1) s += __shfl_xor(s, o, 32);
  if (lane == 0) psi[row] = s;
}

// Time-embedding MLP + its first/second tau derivatives (forward mode),
// pushed through Wf; plus wxs = row-sums of Wx; plus scalar accum zeroing.
__global__ __launch_bounds__(256) void time_k(const float* Wt1, const float* bt1,
                                              const float* Wt2, const float* bt2,
                                              const float* Wf, const float* Wx,
                                              float* te4, float* S2d4, float* S2dd4,
                                              float* wxs, float* scal) {
  __shared__ float e0[D_], e1[D_], e2[D_];
  __shared__ float g0[D_], g1[D_], g2[D_];
  const int tid = threadIdx.x;
  if (blockIdx.x == 4) {
    for (int r = tid; r < D_; r += 256) {
      float s = 0.f;
      for (int d = 0; d < D_; ++d) s += Wx[(size_t)r * D_ + d];
      wxs[r] = s;
    }
    if (tid < 8) scal[tid] = 0.f;   // utt[4], ic, pde, jvp accumulators
    return;
  }
  const float tau = (float)blockIdx.x;
  {
    int j = tid;  // 256 threads, half = 256
    if (j < 256) {
      float f = expf(-9.210340371976184f * (float)j / 255.f);
      float ph = tau * f, sn = sinf(ph), cs = cosf(ph);
      e0[j] = sn;          e0[j + 256] = cs;
      e1[j] = f * cs;      e1[j + 256] = -f * sn;
      e2[j] = -f * f * sn; e2[j + 256] = -f * f * cs;
    }
  }
  __syncthreads();
  for (int r = tid; r < D_; r += 256) {
    float a = bt1[r], ad = 0.f, add = 0.f;
    const float* wr = Wt1 + (size_t)r * D_;
    for (int j = 0; j < D_; ++j) {
      float w = wr[j];
      a += w * e0[j]; ad += w * e1[j]; add += w * e2[j];
    }
    float Phi = 0.5f * (1.f + erff(a * 0.70710678118654752f));
    float phi = 0.3989422804014327f * expf(-0.5f * a * a);
    float gp = Phi + a * phi;            // gelu'
    float gpp = phi * (2.f - a * a);     // gelu''
    g0[r] = a * Phi;
    g1[r] = gp * ad;
    g2[r] = gpp * ad * ad + gp * add;
  }
  __syncthreads();
  for (int r = tid; r < D_; r += 256) {
    float t = bt2[r], td = 0.f, tdd = 0.f;
    const float* wr = Wt2 + (size_t)r * D_;
    for (int j = 0; j < D_; ++j) {
      float w = wr[j];
      t += w * g0[j]; td += w * g1[j]; tdd += w * g2[j];
    }
    te4[blockIdx.x * D_ + r] = t;
    e0[r] = td;   // reuse as te_dot
    e1[r] = tdd;  // reuse as te_ddot
  }
  __syncthreads();
  for (int r = tid; r < D_; r += 256) {
    float sdv = 0.f, sddv = 0.f;
    const float* wr = Wf + (size_t)r * D_;
    for (int j = 0; j < D_; ++j) {
      float w = wr[j];
      sdv += w * e0[j]; sddv += w * e1[j];
    }
    S2d4[blockIdx.x * D_ + r] = sdv;    // s2_dot  (shared across batch)
    S2dd4[blockIdx.x * D_ + r] = sddv;  // s2_ddot
  }
}

__global__ void final_k(const float* scal, float* out) {
  float mse_ic = scal[4] / (float)(B_ * D_);
  float mse_pde = scal[5] / (float)(B_ * D_);
  float mse_jvp = scal[6] / ((float)B_ * (float)DG_);
  out[0] = mse_ic + mse_pde * 0.25f - mse_jvp;
}

extern "C" void kernel_launch(void* const* d_in, const int* in_sizes, int n_in,
                              void* d_out, int out_size, void* d_ws, size_t ws_size,
                              hipStream_t stream) {
  (void)in_sizes; (void)n_in; (void)out_size; (void)ws_size;
  const float* z   = (const float*)d_in[0];
  const float* Wx  = (const float*)d_in[1];
  const float* bx  = (const float*)d_in[2];
  const float* Wt1 = (const float*)d_in[3];
  const float* bt1 = (const float*)d_in[4];
  const float* Wt2 = (const float*)d_in[5];
  const float* bt2 = (const float*)d_in[6];
  const float* Wf  = (const float*)d_in[7];
  const float* bf  = (const float*)d_in[8];
  const float* Wo  = (const float*)d_in[9];
  const float* bo  = (const float*)d_in[10];
  const float* c   = (const float*)d_in[11];
  const float* G   = (const float*)d_in[12];
  const int* idx   = (const int*)d_in[13];
  const int* tsel  = (const int*)d_in[14];

  float* out = (float*)d_out;
  float* ws = (float*)d_ws;
  const size_t MD = (size_t)B_ * D_;

  float* X   = ws;
  float* H   = X + MD;
  float* DX  = H + MD;
  float* Q   = DX + MD;
  float* GAM = Q + MD;
  float* UZ  = GAM + MD;
  float* ZA  = UZ + MD;
  float* ZB  = ZA + MD;
  float* TE   = ZB + MD;       // 4 * 512
  float* S2D  = TE + 4 * D_;   // 4 * 512
  float* S2DD = S2D + 4 * D_;  // 4 * 512
  float* WXS  = S2DD + 4 * D_; // 512
  float* UV   = WXS + D_;      // 2048
  float* R3   = UV + B_;       // 2048
  float* PSI  = R3 + B_;       // 2048
  float* SCAL = PSI + B_;      // 8: utt[0..3], ic, pde, jvp

  float* energy = out;
  float* lat1 = out + B_;
  float* lat2 = lat1 + MD;
  float* lossp = lat2 + MD;

  time_k<<<dim3(5), dim3(256), 0, stream>>>(Wt1, bt1, Wt2, bt2, Wf, Wx,
                                            TE, S2D, S2DD, WXS, SCAL);

  const dim3 blk(256);
  const dim3 gmain(D_ / BN, B_ / BM);   // (8, 32)
  const dim3 gjvp(DG_ / BN, B_ / BM);   // (32, 32)
  const dim3 gred(B_ / 8);

  const float* zc = z;
  float* zn = ZA;
  for (int step = 0; step < 4; ++step) {
    GP base{};
    base.te = TE + step * D_;
    base.x = X; base.h = H; base.dx = DX; base.q = Q;
    base.r3 = R3; base.u = UV; base.psi = PSI;
    base.Wo = Wo; base.wxs = WXS; base.zcur = zc; base.uz = UZ;
    base.cptr = c; base.idxp = idx; base.tselp = tsel; base.scal = SCAL;
    base.znext = zn; base.lat1 = lat1; base.lat2 = lat2;
    base.step = step; base.N = D_;

    { GP p = base; p.Ain = zc; p.W = Wx; p.bias = bx; p.Out = X;
      gemm_k<0><<<gmain, blk, 0, stream>>>(p); }                       // x
    { GP p = base; p.Ain = X; p.W = Wf; p.bias = bf; p.Out = H;
      gemm_k<1><<<gmain, blk, 0, stream>>>(p); }                       // h
    r1_k<<<gred, blk, 0, stream>>>(H, Wo, bo, S2D + step * D_, S2DD + step * D_,
                                   UV, R3, SCAL + step, energy, tsel, step);
    { GP p = base; p.Ain = H; p.W = Wf; p.Out = DX;
      gemm_k<2><<<gmain, blk, 0, stream>>>(p); }                       // d_x
    { GP p = base; p.Ain = DX; p.W = Wx; p.Out = UZ; p.accum = SCAL + 4;
      gemm_k<3><<<gmain, blk, 0, stream>>>(p); }                       // u_z, z+, latents, ic
    { GP p = base; p.Ain = UZ; p.W = G; p.accum = SCAL + 6; p.N = DG_;
      gemm_k<7><<<gjvp, blk, 0, stream>>>(p); }                        // jvp (self-guards)
    { GP p = base; p.Ain = X; p.W = Wf; p.Out = Q;
      gemm_k<4><<<gmain, blk, 0, stream>>>(p); }                       // q
    r2_k<<<gred, blk, 0, stream>>>(H, Wo, Q, PSI);                     // psi
    { GP p = base; p.Ain = H; p.W = Wf; p.Out = GAM;
      gemm_k<5><<<gmain, blk, 0, stream>>>(p); }                       // gamma
    { GP p = base; p.Ain = GAM; p.W = Wx; p.accum = SCAL + 5;
      gemm_k<6><<<gmain, blk, 0, stream>>>(p); }                       // u_zz -> pde sum
    zc = zn;
    zn = (zn == ZA) ? ZB : ZA;
  }
  final_k<<<1, 1, 0, stream>>>(SCAL, lossp);
}